// RelationalAwareEncoder_63153199120592
// MI455X (gfx1250) — compile-verified
//
#include <hip/hip_runtime.h>

typedef __attribute__((ext_vector_type(2))) float v2f;
typedef __attribute__((ext_vector_type(8))) float v8f;

namespace {
constexpr int       kN       = 100000;
constexpr int       kE       = 50000;
constexpr long long kNNZ     = 3200000LL;
constexpr int       kD       = 32;
constexpr float     kAlpha   = 0.5f;
constexpr int       kRowTiles = kN / 16;          // 6250 (exact)
constexpr int       kChunk    = 512;              // incidences per wave in NNZ passes
constexpr int       kWPB      = 8;                // waves per block (wave32)
constexpr int       kBT       = 256;              // block threads
static_assert(kN % 16 == 0, "row tiles exact");
static_assert(kNNZ % kChunk == 0, "chunks exact");
}

// ---- one 16x16 (M,N) x K=4 fp32 WMMA step ------------------------------
__device__ __forceinline__ v8f wmma4(v2f a, v2f b, v8f c) {
  // 8 args: (neg_a, A, neg_b, B, c_mod, C, reuse_a, reuse_b)
  return __builtin_amdgcn_wmma_f32_16x16x4_f32(false, a, false, b, (short)0, c,
                                               false, false);
}

// ---- k0: zero Xe / Acc / deg (contiguous region of workspace) ----------
__global__ void k0_zero(float* __restrict__ p, size_t n) {
  size_t i = (size_t)blockIdx.x * blockDim.x + threadIdx.x;
  size_t stride = (size_t)gridDim.x * blockDim.x;
  for (; i < n; i += stride) p[i] = 0.0f;
}

// ---- k1: X1 = X@W1 + b1 ; XA = X@W2[:32] (shared A fragments) ----------
__global__ void k1_dual_gemm(const float* __restrict__ X,
                             const float* __restrict__ W1,
                             const float* __restrict__ b1,
                             const float* __restrict__ W2,   // [64 x 32], rows 0..31 used
                             float* __restrict__ X1,
                             float* __restrict__ XA) {
  int wave = (int)((blockIdx.x * blockDim.x + threadIdx.x) >> 5);
  int lane = (int)(threadIdx.x & 31);
  if (wave >= kRowTiles) return;                 // wave-uniform: EXEC stays all-1s
  const int m0 = wave * 16;
  const int m  = lane & 15;
  const int kh = (lane >> 4) * 2;                // lanes 16-31 hold K+2,K+3
  const float* Arow = X + (size_t)(m0 + m) * kD;

  v2f a[8];
#pragma unroll
  for (int s = 0; s < 8; ++s) {
    int k = 4 * s + kh;
    a[s].x = Arow[k];
    a[s].y = Arow[k + 1];
  }

#pragma unroll
  for (int half = 0; half < 2; ++half) {
    const int n0 = half * 16;
    v8f c1 = {};
    v8f c2 = {};
#pragma unroll
    for (int s = 0; s < 8; ++s) {
      int k = 4 * s + kh;
      v2f bw1, bw2;
      bw1.x = W1[(size_t)k * kD + n0 + m];
      bw1.y = W1[(size_t)(k + 1) * kD + n0 + m];
      bw2.x = W2[(size_t)k * kD + n0 + m];
      bw2.y = W2[(size_t)(k + 1) * kD + n0 + m];
      c1 = wmma4(a[s], bw1, c1);
      c2 = wmma4(a[s], bw2, c2);
    }
    const float bias = b1[n0 + m];
#pragma unroll
    for (int j = 0; j < 8; ++j) {
      int row = m0 + j + ((lane >> 4) << 3);
      int col = n0 + m;
      X1[(size_t)row * kD + col] = c1[j] + bias;
      XA[(size_t)row * kD + col] = c2[j];
    }
  }
}

// ---- k2: Xe[e] += X1[vertex[i]] * atts[i] over sorted edges ------------
// Sorted runs (avg length 64) accumulate in registers; atomics only at
// run boundaries. lane = feature index. Indices loaded coalesced + __shfl.
__global__ void k2_edge_agg(const float* __restrict__ X1,
                            const float* __restrict__ atts,
                            const int* __restrict__ vertex,
                            const int* __restrict__ edges,
                            float* __restrict__ Xe) {
  int wave = (int)((blockIdx.x * blockDim.x + threadIdx.x) >> 5);
  int lane = (int)(threadIdx.x & 31);
  long long i0 = (long long)wave * kChunk;
  if (i0 >= kNNZ) return;
  long long i1 = i0 + kChunk;
  if (i1 > kNNZ) i1 = kNNZ;

  int   cur = edges[i0];
  float acc = 0.0f;
  for (long long base = i0; base < i1; base += 32) {
    if (base + 1024 < kNNZ) {
      __builtin_prefetch(&edges[base + 1024], 0, 1);
      __builtin_prefetch(&vertex[base + 1024], 0, 1);
    }
    long long idx = base + lane;
    bool ok = idx < i1;
    int   eL = ok ? edges[idx]  : 0;
    int   vL = ok ? vertex[idx] : 0;
    float aL = ok ? atts[idx]   : 0.0f;
    int nv = (int)(((i1 - base) < 32) ? (i1 - base) : 32);
    for (int t = 0; t < nv; ++t) {
      int   e = __shfl(eL, t, 32);
      int   v = __shfl(vL, t, 32);
      float w = __shfl(aL, t, 32);
      float val = X1[(size_t)v * kD + lane] * w;
      if (e != cur) {
        atomicAdd(&Xe[(size_t)cur * kD + lane], acc);
        cur = e;
        acc = val;
      } else {
        acc += val;
      }
    }
  }
  atomicAdd(&Xe[(size_t)cur * kD + lane], acc);
}

// ---- k3: Acc[vertex[i]] += Xe[edges[i]] ; deg[vertex[i]] += 1 ----------
__global__ void k3_scatter(const float* __restrict__ Xe,
                           const int* __restrict__ vertex,
                           const int* __restrict__ edges,
                           float* __restrict__ Acc,
                           float* __restrict__ deg) {
  int wave = (int)((blockIdx.x * blockDim.x + threadIdx.x) >> 5);
  int lane = (int)(threadIdx.x & 31);
  long long i0 = (long long)wave * kChunk;
  if (i0 >= kNNZ) return;
  long long i1 = i0 + kChunk;
  if (i1 > kNNZ) i1 = kNNZ;

  for (long long base = i0; base < i1; base += 32) {
    if (base + 1024 < kNNZ) {
      __builtin_prefetch(&vertex[base + 1024], 0, 1);
      __builtin_prefetch(&edges[base + 1024], 0, 1);
    }
    long long idx = base + lane;
    bool ok = idx < i1;
    int eL = ok ? edges[idx]  : 0;
    int vL = ok ? vertex[idx] : 0;
    int nv = (int)(((i1 - base) < 32) ? (i1 - base) : 32);
    for (int t = 0; t < nv; ++t) {
      int e = __shfl(eL, t, 32);
      int v = __shfl(vL, t, 32);
      atomicAdd(&Acc[(size_t)v * kD + lane], Xe[(size_t)e * kD + lane]);
      if (lane == 0) atomicAdd(&deg[v], 1.0f);
    }
  }
}

// ---- k4: T = Acc@W2[32:]; Xv = deg*XA + T + deg*b2;
//          Xr = (1-a)Xv + a*X0; out = Xr@W + b  (WMMA, LDS-staged) -------
__global__ void k4_out(const float* __restrict__ Acc,
                       const float* __restrict__ XA,
                       const float* __restrict__ deg,
                       const float* __restrict__ X0,
                       const float* __restrict__ W2,   // [64x32], rows 32..63 used
                       const float* __restrict__ b2,
                       const float* __restrict__ W,
                       const float* __restrict__ b,
                       float* __restrict__ out) {
  __shared__ float xr[kWPB][16][kD + 1];
  int wid  = (int)(threadIdx.x >> 5);
  int lane = (int)(threadIdx.x & 31);
  int wave = (int)blockIdx.x * kWPB + wid;
  if (wave >= kRowTiles) return;
  const int m0 = wave * 16;
  const int m  = lane & 15;
  const int kh = (lane >> 4) * 2;
  const float* W2b = W2 + (size_t)32 * kD;

  // A fragments from Acc tile
  const float* Arow = Acc + (size_t)(m0 + m) * kD;
  v2f a[8];
#pragma unroll
  for (int s = 0; s < 8; ++s) {
    int k = 4 * s + kh;
    a[s].x = Arow[k];
    a[s].y = Arow[k + 1];
  }

#pragma unroll
  for (int half = 0; half < 2; ++half) {
    const int n0 = half * 16;
    v8f c = {};
#pragma unroll
    for (int s = 0; s < 8; ++s) {
      int k = 4 * s + kh;
      v2f bw;
      bw.x = W2b[(size_t)k * kD + n0 + m];
      bw.y = W2b[(size_t)(k + 1) * kD + n0 + m];
      c = wmma4(a[s], bw, c);
    }
    const float b2n = b2[n0 + m];
#pragma unroll
    for (int j = 0; j < 8; ++j) {
      int row  = j + ((lane >> 4) << 3);
      int grow = m0 + row;
      int col  = n0 + m;
      float dg = deg[grow];
      float xv = dg * XA[(size_t)grow * kD + col] + c[j] + dg * b2n;
      xr[wid][row][col] =
          (1.0f - kAlpha) * xv + kAlpha * X0[(size_t)grow * kD + col];
    }
  }

  // second GEMM: out = Xr @ W + b (A fragments re-read from LDS; same-wave
  // LDS ops are in-order, compiler inserts s_wait_dscnt before use)
  v2f a2[8];
#pragma unroll
  for (int s = 0; s < 8; ++s) {
    int k = 4 * s + kh;
    a2[s].x = xr[wid][m][k];
    a2[s].y = xr[wid][m][k + 1];
  }
#pragma unroll
  for (int half = 0; half < 2; ++half) {
    const int n0 = half * 16;
    v8f c = {};
#pragma unroll
    for (int s = 0; s < 8; ++s) {
      int k = 4 * s + kh;
      v2f bw;
      bw.x = W[(size_t)k * kD + n0 + m];
      bw.y = W[(size_t)(k + 1) * kD + n0 + m];
      c = wmma4(a2[s], bw, c);
    }
    const float bn = b[n0 + m];
#pragma unroll
    for (int j = 0; j < 8; ++j) {
      int row = m0 + j + ((lane >> 4) << 3);
      int col = n0 + m;
      out[(size_t)row * kD + col] = c[j] + bn;
    }
  }
}

extern "C" void kernel_launch(void* const* d_in, const int* in_sizes, int n_in,
                              void* d_out, int out_size, void* d_ws, size_t ws_size,
                              hipStream_t stream) {
  (void)in_sizes; (void)n_in; (void)out_size; (void)ws_size;
  const float* X    = (const float*)d_in[0];
  const float* X0   = (const float*)d_in[1];
  const float* atts = (const float*)d_in[2];
  const float* W1w  = (const float*)d_in[3];
  const float* W1b  = (const float*)d_in[4];
  const float* W2w  = (const float*)d_in[5];
  const float* W2bv = (const float*)d_in[6];
  const float* Ww   = (const float*)d_in[7];
  const float* Wbv  = (const float*)d_in[8];
  const int*   vertex = (const int*)d_in[9];
  const int*   edges  = (const int*)d_in[10];
  float* out = (float*)d_out;

  // workspace layout (floats): X1 | XA | Xe | Acc | deg  (~45.2 MB)
  float* ws  = (float*)d_ws;
  float* X1  = ws;
  float* XA  = X1 + (size_t)kN * kD;
  float* Xe  = XA + (size_t)kN * kD;
  float* Acc = Xe + (size_t)kE * kD;
  float* dg  = Acc + (size_t)kN * kD;

  const size_t zeroN = (size_t)kE * kD + (size_t)kN * kD + (size_t)kN;
  k0_zero<<<512, kBT, 0, stream>>>(Xe, zeroN);

  const int gemmBlocks = (kRowTiles + kWPB - 1) / kWPB;
  k1_dual_gemm<<<gemmBlocks, kBT, 0, stream>>>(X, W1w, W1b, W2w, X1, XA);

  const int aggWaves  = (int)((kNNZ + kChunk - 1) / kChunk);
  const int aggBlocks = (aggWaves + kWPB - 1) / kWPB;
  k2_edge_agg<<<aggBlocks, kBT, 0, stream>>>(X1, atts, vertex, edges, Xe);
  k3_scatter<<<aggBlocks, kBT, 0, stream>>>(Xe, vertex, edges, Acc, dg);

  k4_out<<<gemmBlocks, kBT, 0, stream>>>(Acc, XA, dg, X0, W2w, W2bv, Ww, Wbv, out);
}